// MEF_MSSSIM_49907519979848
// MI455X (gfx1250) — compile-verified
//
#include <hip/hip_runtime.h>
#include <math.h>

typedef float v2f __attribute__((ext_vector_type(2)));
typedef float v8f __attribute__((ext_vector_type(8)));

#define IMG_H 1024
#define IMG_W 1024
#define WAVES_PER_WG 2
#define TILE_H 16
#define TILE_W 32                 // per-wave output region: 16 x 32
#define TILES_X (IMG_W / TILE_W)  // 32
#define TILES_PER_IMG (TILES_X * (IMG_H / TILE_H))  // 2048
#define PATCH_COLS 42             // 32 + 10 halo (valid data cols)
#define PATCH_CSTRIDE 48          // cols stored (42 data + 6 zero)
#define PATCH_RSTRIDE 30          // rows per col: 26 data + 4 zero (26..29)
#define PATCH_ELEMS (PATCH_CSTRIDE * PATCH_RSTRIDE)  // 1440
#define TMP_STRIDE 50             // 16 x 50 tmp, conflict-free, pair-aligned
#define C2F 9.0e-4f               // 0.03^2

// ---------------------------------------------------------------------------
// WMMA wrapper: D = A(16x4 f32) * B(4x16 f32) + C  (full fp32 matrix op)
// ---------------------------------------------------------------------------
__device__ __forceinline__ v8f wmma_f32(v2f a, v2f b, v8f c) {
  return __builtin_amdgcn_wmma_f32_16x16x4_f32(false, a, false, b, (short)0, c,
                                               false, false);
}

// ---------------------------------------------------------------------------
// Fill a column-major zero-padded patch: dst[col*30 + row].
// Rows y0-5..y0+20 (26), cols x0-5..x0+36 (42); rows 26..29 and cols 42..47
// zeroed.  Global reads coalesced along cols; LDS writes are paired b64.
// ---------------------------------------------------------------------------
__device__ __forceinline__ void fill_patch_t(const float* __restrict__ src,
                                             float* dst, int y0, int x0,
                                             int lane) {
#pragma unroll
  for (int ph = 0; ph < 2; ++ph) {
    const int col = ph * 32 + lane;
    if (col < PATCH_CSTRIDE) {     // ph 1: lanes 16..31 idle (no WMMA inside)
      const int gx = x0 - 5 + col;
      const bool inx = (col < PATCH_COLS) && (gx >= 0) && (gx < IMG_W);
      const int cx = gx < 0 ? 0 : (gx > IMG_W - 1 ? IMG_W - 1 : gx);
      float* dcol = dst + col * PATCH_RSTRIDE;
#pragma unroll
      for (int r = 0; r < 26; r += 2) {
        v2f pv;
#pragma unroll
        for (int u = 0; u < 2; ++u) {
          int gy = y0 - 5 + r + u;
          bool iny = (gy >= 0) && (gy < IMG_H);
          int cy = gy < 0 ? 0 : (gy > IMG_H - 1 ? IMG_H - 1 : gy);
          float v = src[(size_t)cy * IMG_W + cx];
          pv[u] = (inx && iny) ? v : 0.0f;
        }
        *(v2f*)(dcol + r) = pv;
      }
      v2f z = {0.0f, 0.0f};
      *(v2f*)(dcol + 26) = z;
      *(v2f*)(dcol + 28) = z;
    }
  }
}

// ---------------------------------------------------------------------------
// Separable 11x11 Gaussian blur over a 16x32 region on the WMMA pipe.
// `pair(off)` returns plane values {row, row+1} at patch offset (col-major,
// adjacent dwords -> one ds_load_b64 straight into the operand pair).
// Pass 1: tmp(16x48) = Wv * P   -> 21 wmma ; Pass 2: out_t = tmp * Wh -> 14.
// Weight pairs come pre-packed per lane from wtab (aligned b64 loads) and are
// reused as pass-1 A and pass-2 B (identical per-lane layout).
// ---------------------------------------------------------------------------
template <typename PV>
__device__ __forceinline__ void blur_tile(const float* wtab, int lane,
                                          float* sTmpw, PV pair, v8f& out0,
                                          v8f& out1) {
  const int hi = lane >> 4, lo = lane & 15;
  const int pb = lo * PATCH_RSTRIDE + 2 * hi;    // patch base (col=lo,row=2hi)
  v2f aw[7];
#pragma unroll
  for (int j = 0; j < 7; ++j)
    aw[j] = *(const v2f*)(wtab + (j * 32 + lane) * 2);

  v8f acc0 = (v8f){0.f, 0.f, 0.f, 0.f, 0.f, 0.f, 0.f, 0.f};
  v8f acc1 = acc0, acc2 = acc0;
#pragma unroll
  for (int j = 0; j < 7; ++j) {                  // rows 4j+2hi, +1 (26..27=0)
    acc0 = wmma_f32(aw[j], pair(pb + 4 * j), acc0);
    acc1 = wmma_f32(aw[j], pair(pb + 16 * PATCH_RSTRIDE + 4 * j), acc1);
    acc2 = wmma_f32(aw[j], pair(pb + 32 * PATCH_RSTRIDE + 4 * j), acc2);
  }
  const int tb = (8 * hi) * TMP_STRIDE + lo;     // tmp row-major 16x50
#pragma unroll
  for (int d = 0; d < 8; ++d) {
    sTmpw[tb + d * TMP_STRIDE] = acc0[d];
    sTmpw[tb + d * TMP_STRIDE + 16] = acc1[d];
    sTmpw[tb + d * TMP_STRIDE + 32] = acc2[d];
  }
  asm volatile("" ::: "memory");  // same-wave LDS RAW: keep load after store
  const int ab = lo * TMP_STRIDE + 2 * hi;
  v8f o0 = (v8f){0.f, 0.f, 0.f, 0.f, 0.f, 0.f, 0.f, 0.f};
  v8f o1 = o0;
#pragma unroll
  for (int j = 0; j < 7; ++j) {
    v2f a0 = *(const v2f*)(sTmpw + ab + 4 * j);        // b64 load
    v2f a1 = *(const v2f*)(sTmpw + ab + 16 + 4 * j);   // b64 load
    o0 = wmma_f32(a0, aw[j], o0);
    o1 = wmma_f32(a1, aw[j], o1);
  }
  asm volatile("" ::: "memory");  // WAR vs next plane's tmp stores
  out0 = o0;                                     // cols x0..x0+15
  out1 = o1;                                     // cols x0+16..x0+31
}

// ---------------------------------------------------------------------------
// Main kernel: one wave per 16x32 region, 2 waves per workgroup.
// ---------------------------------------------------------------------------
__global__ __launch_bounds__(64) void mef_msssim_main(
    const float* __restrict__ X, const float* __restrict__ Ys,
    float* __restrict__ partial) {
  __shared__ float sX[WAVES_PER_WG][PATCH_ELEMS];
  __shared__ float sY[WAVES_PER_WG][3][PATCH_ELEMS];
  __shared__ float sTmp[WAVES_PER_WG][16 * TMP_STRIDE];
  __shared__ float wtab[7 * 32 * 2];             // per-lane weight pairs
  __shared__ float gtab[16];                     // 1D gaussian taps + zeros
  __shared__ float sRed[WAVES_PER_WG];

  const int tid = threadIdx.x;
  const int w = tid >> 5;
  const int lane = tid & 31;
  const int hi = lane >> 4, lo = lane & 15;

  const int tile = blockIdx.x * WAVES_PER_WG + w;
  const int k = tile >> 11;                      // / TILES_PER_IMG (2048)
  const int rem = tile & (TILES_PER_IMG - 1);
  const int y0 = (rem >> 5) * TILE_H;            // rem / TILES_X
  const int x0 = (rem & (TILES_X - 1)) * TILE_W;

  // Gaussian window (double, matching reference float64 window computation;
  // constant-folded to f32 literals by the compiler)
  double gs[11];
  double ssum = 0.0;
#pragma unroll
  for (int i = 0; i < 11; ++i) {
    double x = (double)(i - 5);
    double s = 11.0 / 6.0;
    gs[i] = exp(-(x * x) / (2.0 * s * s));
    ssum += gs[i];
  }
  float g1[11];
#pragma unroll
  for (int i = 0; i < 11; ++i) g1[i] = (float)(gs[i] / ssum);

  if (tid < 16) {
    float v = 0.0f;
#pragma unroll
    for (int t = 0; t < 11; ++t)
      if (tid == t) v = g1[t];
    gtab[tid] = v;
  }

  // stage raw patches (column-major, zero-padded)
  const float* Xk = X + (size_t)k * (IMG_H * IMG_W);
  fill_patch_t(Xk, sX[w], y0, x0, lane);
#pragma unroll
  for (int c = 0; c < 3; ++c)
    fill_patch_t(Ys + ((size_t)k * 3 + c) * (IMG_H * IMG_W), sY[w][c], y0, x0,
                 lane);
  __syncthreads();

  // band-matrix weight pairs -> LDS (lane-dependent only; wave 0 publishes)
  if (w == 0) {
#pragma unroll
    for (int j = 0; j < 7; ++j) {
      v2f aw;
#pragma unroll
      for (int v = 0; v < 2; ++v) {
        int d = 4 * j + 2 * hi + v - lo;         // tap index, valid in [0,10]
        int dc = d < 0 ? 0 : (d > 15 ? 15 : d);
        float t = gtab[dc];                      // gtab[11..15] == 0
        aw[v] = (d >= 0 && d < 16) ? t : 0.0f;
      }
      *(v2f*)(wtab + (j * 32 + lane) * 2) = aw;
    }
  }
  __syncthreads();

  const float* px = sX[w];
  float* tw = sTmp[w];

  auto pvX = [&](int o) { return *(const v2f*)(px + o); };
  auto pvX2 = [&](int o) {
    v2f x = *(const v2f*)(px + o);
    return x * x;
  };

  // fused-image stats (two 16x16 sub-tiles)
  v8f rMuX[2], rEX2[2];
  blur_tile(wtab, lane, tw, pvX, rMuX[0], rMuX[1]);
  blur_tile(wtab, lane, tw, pvX2, rEX2[0], rEX2[1]);

  float muXv[2][8], dpart[2][8];
#pragma unroll
  for (int t = 0; t < 2; ++t)
#pragma unroll
    for (int d = 0; d < 8; ++d) {
      muXv[t][d] = rMuX[t][d];
      dpart[t][d] = (rEX2[t][d] - muXv[t][d] * muXv[t][d]) + C2F;  // sigX2+C2
    }

  // per-channel stats folded into running argmax state (rc kept as sn/sd so
  // the argmax uses a cross-multiply compare -- no divisions)
  float sn[2][8], sdn[2][8], cssel[2][8];
#pragma unroll
  for (int c = 0; c < 3; ++c) {
    const float* py = sY[w][c];
    auto pvY = [&](int o) { return *(const v2f*)(py + o); };
    auto pvY2 = [&](int o) {
      v2f y = *(const v2f*)(py + o);
      return y * y;
    };
    auto pvXY = [&](int o) {
      return (*(const v2f*)(px + o)) * (*(const v2f*)(py + o));
    };
    v8f rMuY[2], rEY2[2], rEXY[2];
    blur_tile(wtab, lane, tw, pvY, rMuY[0], rMuY[1]);
    blur_tile(wtab, lane, tw, pvY2, rEY2[0], rEY2[1]);
    blur_tile(wtab, lane, tw, pvXY, rEXY[0], rEXY[1]);
#pragma unroll
    for (int t = 0; t < 2; ++t)
#pragma unroll
      for (int d = 0; d < 8; ++d) {
        float mu = rMuY[t][d];
        float m2 = mu * mu;
        float s2 = rEY2[t][d] - m2;
        float den = fmaxf(m2, 0.2f);             // rc = s2/den (not divided)
        float sigXY = rEXY[t][d] - muXv[t][d] * mu;
        float cs =
            (2.0f * sigXY + C2F) * __builtin_amdgcn_rcpf(dpart[t][d] + s2);
        bool take = (c == 0) || (s2 * sdn[t][d] > sn[t][d] * den);
        sn[t][d] = take ? s2 : sn[t][d];
        sdn[t][d] = take ? den : sdn[t][d];
        cssel[t][d] = take ? cs : cssel[t][d];
      }
  }

  float lsum = 0.0f;
#pragma unroll
  for (int t = 0; t < 2; ++t)
#pragma unroll
    for (int d = 0; d < 8; ++d) lsum += cssel[t][d];

  // wave reduce, then workgroup partial (no atomics -> deterministic)
#pragma unroll
  for (int off = 16; off > 0; off >>= 1) lsum += __shfl_down(lsum, off, 32);
  if (lane == 0) sRed[w] = lsum;
  __syncthreads();
  if (tid == 0) partial[blockIdx.x] = sRed[0] + sRed[1];
}

// ---------------------------------------------------------------------------
// Deterministic final reduction of per-WG partials -> mean scalar
// ---------------------------------------------------------------------------
__global__ void mef_msssim_reduce(const float* __restrict__ partial, int n,
                                  float inv_count, float* __restrict__ out) {
  __shared__ float s[256];
  float a = 0.0f;
  for (int i = threadIdx.x; i < n; i += 256) a += partial[i];
  s[threadIdx.x] = a;
  __syncthreads();
  for (int off = 128; off > 0; off >>= 1) {
    if ((int)threadIdx.x < off) s[threadIdx.x] += s[threadIdx.x + off];
    __syncthreads();
  }
  if (threadIdx.x == 0) out[0] = s[0] * inv_count;
}

extern "C" void kernel_launch(void* const* d_in, const int* in_sizes, int n_in,
                              void* d_out, int out_size, void* d_ws,
                              size_t ws_size, hipStream_t stream) {
  const float* X = (const float*)d_in[0];    // [K,1,1024,1024]
  const float* Ys = (const float*)d_in[1];   // [K,3,1024,1024]
  float* partial = (float*)d_ws;

  const int K = in_sizes[0] / (IMG_H * IMG_W);        // 8
  const int nwaves = K * TILES_PER_IMG;               // 16384
  const int nwg = nwaves / WAVES_PER_WG;              // 8192

  mef_msssim_main<<<nwg, 64, 0, stream>>>(X, Ys, partial);
  const float inv_count = 1.0f / ((float)K * IMG_H * IMG_W);
  mef_msssim_reduce<<<1, 256, 0, stream>>>(partial, nwg, inv_count,
                                           (float*)d_out);
}